// GraphTransformer_38233798869666
// MI455X (gfx1250) — compile-verified
//
#include <hip/hip_runtime.h>

typedef __attribute__((ext_vector_type(2))) float v2f;
typedef __attribute__((ext_vector_type(8))) float v8f;

constexpr int N_    = 50000;
constexpr int E_    = 1600000;
constexpr int F_    = 64;
constexpr int IN_   = 8;
constexpr int BANDS_= 8;
constexpr int HC_   = 32;
constexpr int GIN_  = 256;   // BANDS_*HC_
constexpr int OUT_  = 32;

// ---------------------------------------------------------------- init
__global__ void k_init(float* __restrict__ m, float* __restrict__ z,
                       float* __restrict__ deg) {
    int tid = blockIdx.x * blockDim.x + threadIdx.x;
    if (tid < N_ * 16) {
        m[tid] = __int_as_float(0xFF800000);  // -inf
        z[tid] = 0.0f;
    }
    if (tid < N_) deg[tid] = 2.0f;            // self-loop weight
}

// ---------------------------------------------------- node projections (WMMA)
// Per wave: one 16-node tile. out16x32 = X16x8 @ W8x32 (+bias), per band,
// for q,k,v,skip. K=8 -> two K=4 WMMA steps; HC=32 -> two 16-col N tiles.
// Weights staged once per block in LDS (4x256 floats + 4x32 biases).
__global__ void k_proj(const float* __restrict__ x,
                       const float* __restrict__ Wq, const float* __restrict__ bq,
                       const float* __restrict__ Wk, const float* __restrict__ bk,
                       const float* __restrict__ Wv, const float* __restrict__ bv,
                       const float* __restrict__ Wsk,const float* __restrict__ bsk,
                       float* __restrict__ qb, float* __restrict__ kb,
                       float* __restrict__ vb, float* __restrict__ hb) {
    __shared__ float sW[4 * 256];
    __shared__ float sB[4 * 32];
    {
        const int t = threadIdx.x;                 // 256 threads
        sW[0 * 256 + t] = Wq[t];
        sW[1 * 256 + t] = Wk[t];
        sW[2 * 256 + t] = Wv[t];
        sW[3 * 256 + t] = Wsk[t];
        if (t < 32) {
            sB[0 * 32 + t] = bq[t];
            sB[1 * 32 + t] = bk[t];
            sB[2 * 32 + t] = bv[t];
            sB[3 * 32 + t] = bsk[t];
        }
    }
    __syncthreads();

    const int wave = threadIdx.x >> 5;
    const int lane = threadIdx.x & 31;
    const int tile = blockIdx.x * (blockDim.x >> 5) + wave;
    if (tile * 16 >= N_) return;              // wave-uniform (after barrier)
    const int base = tile * 16;
    const int half = lane >> 4;               // 0: K=0,1 ; 1: K=2,3
    const int l    = lane & 15;

    float* Os[4] = {qb, kb, vb, hb};

    const float* xrow = x + (size_t)(base + l) * F_;

    for (int b = 0; b < BANDS_; ++b) {
        // A tiles (16x4) for k0=0 and k0=4
        v2f A0, A1;
        A0.x = xrow[b * IN_ + 2 * half + 0];
        A0.y = xrow[b * IN_ + 2 * half + 1];
        A1.x = xrow[b * IN_ + 4 + 2 * half + 0];
        A1.y = xrow[b * IN_ + 4 + 2 * half + 1];
#pragma unroll
        for (int mtx = 0; mtx < 4; ++mtx) {
            const float* W = &sW[mtx * 256];
#pragma unroll
            for (int nt = 0; nt < 2; ++nt) {
                const int n = nt * 16 + l;
                v2f B0, B1;                    // B 4x16: lane=col, vgprs=K
                B0.x = W[(2 * half + 0) * HC_ + n];
                B0.y = W[(2 * half + 1) * HC_ + n];
                B1.x = W[(4 + 2 * half + 0) * HC_ + n];
                B1.y = W[(4 + 2 * half + 1) * HC_ + n];
                v8f c = {};
                c = __builtin_amdgcn_wmma_f32_16x16x4_f32(false, A0, false, B0,
                                                          (short)0, c, false, false);
                c = __builtin_amdgcn_wmma_f32_16x16x4_f32(false, A1, false, B1,
                                                          (short)0, c, false, false);
                const float bn = sB[mtx * 32 + n];
                float* O = Os[mtx];
#pragma unroll
                for (int i = 0; i < 8; ++i) {
                    const int row = base + i + 8 * half;   // C/D lane layout
                    O[(size_t)row * GIN_ + b * HC_ + n] = c[i] + bn;
                }
            }
        }
    }
}

// -------------------------------------------- edge pass 1: logits + seg-max
// (also accumulates the GCN degree, once per edge on band 0)
__global__ void k_edge_logits(const int* __restrict__ ei,
                              const float* __restrict__ ea,
                              const float* __restrict__ We,
                              const float* __restrict__ qb,
                              const float* __restrict__ kb,
                              float* __restrict__ logits,
                              float* __restrict__ m,
                              float* __restrict__ deg) {
    int tid = blockIdx.x * blockDim.x + threadIdx.x;
    if (tid >= E_ * BANDS_) return;
    const int e = tid >> 3;
    const int b = tid & 7;
    const int s = ei[e];
    const int d = ei[E_ + e];
    const float ea0 = ea[2 * e], ea1 = ea[2 * e + 1];

    if (b == 0) atomicAdd(&deg[d], ea1);      // GCN degree, once per edge

    const float4* q4 = (const float4*)(qb + (size_t)d * GIN_ + b * HC_);
    const float4* k4 = (const float4*)(kb + (size_t)s * GIN_ + b * HC_);
#pragma unroll
    for (int hh = 0; hh < 2; ++hh) {
        float acc = 0.0f;
#pragma unroll
        for (int j = 0; j < 4; ++j) {          // 4 x float4 = 16 channels
            const float4 qv = q4[hh * 4 + j];
            const float4 kv = k4[hh * 4 + j];
            const int c0 = hh * 16 + j * 4;
            acc += qv.x * (kv.x + ea0 * We[c0 + 0] + ea1 * We[HC_ + c0 + 0]);
            acc += qv.y * (kv.y + ea0 * We[c0 + 1] + ea1 * We[HC_ + c0 + 1]);
            acc += qv.z * (kv.z + ea0 * We[c0 + 2] + ea1 * We[HC_ + c0 + 2]);
            acc += qv.w * (kv.w + ea0 * We[c0 + 3] + ea1 * We[HC_ + c0 + 3]);
        }
        const float lg = acc * 0.25f;          // 1/sqrt(C=16)
        logits[(size_t)e * 16 + b * 2 + hh] = lg;
        float* addr = m + (size_t)d * 16 + b * 2 + hh;
        if (lg >= 0.0f) atomicMax((int*)addr, __float_as_int(lg));
        else atomicMin((unsigned int*)addr, (unsigned int)__float_as_int(lg));
    }
}

// ----------------------------------------- edge pass 2: p = exp(l-m), seg-sum
__global__ void k_edge_exp(const int* __restrict__ ei,
                           float* __restrict__ logits,
                           const float* __restrict__ m,
                           float* __restrict__ z) {
    int tid = blockIdx.x * blockDim.x + threadIdx.x;
    if (tid >= E_ * 16) return;
    const int e   = tid >> 4;
    const int idx = tid & 15;
    const int d   = ei[E_ + e];
    const float p = __expf(logits[tid] - m[(size_t)d * 16 + idx]);
    logits[tid] = p;                            // reuse buffer: now holds p
    atomicAdd(&z[(size_t)d * 16 + idx], p);
}

// ------------------------------------- edge pass 3: weighted message scatter
__global__ void k_edge_agg(const int* __restrict__ ei,
                           const float* __restrict__ ea,
                           const float* __restrict__ We,
                           const float* __restrict__ vb,
                           const float* __restrict__ logits,
                           const float* __restrict__ z,
                           float* __restrict__ hb) {
    int tid = blockIdx.x * blockDim.x + threadIdx.x;
    if (tid >= E_ * BANDS_) return;
    const int e = tid >> 3;
    const int b = tid & 7;
    const int s = ei[e];
    const int d = ei[E_ + e];
    const float ea0 = ea[2 * e], ea1 = ea[2 * e + 1];
    const float a0 = logits[(size_t)e * 16 + b * 2 + 0] /
                     (z[(size_t)d * 16 + b * 2 + 0] + 1e-16f);
    const float a1 = logits[(size_t)e * 16 + b * 2 + 1] /
                     (z[(size_t)d * 16 + b * 2 + 1] + 1e-16f);
    const float4* v4 = (const float4*)(vb + (size_t)s * GIN_ + b * HC_);
    float*        hd = hb + (size_t)d * GIN_ + b * HC_;
#pragma unroll
    for (int j = 0; j < 8; ++j) {              // 8 x float4 = 32 channels
        const float4 vv = v4[j];
        const int c0 = j * 4;
        const float al = (c0 < 16) ? a0 : a1;  // head boundary is float4-aligned
        atomicAdd(&hd[c0 + 0], (vv.x + ea0 * We[c0 + 0] + ea1 * We[HC_ + c0 + 0]) * al);
        atomicAdd(&hd[c0 + 1], (vv.y + ea0 * We[c0 + 1] + ea1 * We[HC_ + c0 + 1]) * al);
        atomicAdd(&hd[c0 + 2], (vv.z + ea0 * We[c0 + 2] + ea1 * We[HC_ + c0 + 2]) * al);
        atomicAdd(&hd[c0 + 3], (vv.w + ea0 * We[c0 + 3] + ea1 * We[HC_ + c0 + 3]) * al);
    }
}

// ------------------------------------------------------------------- ELU
__global__ void k_elu(float* __restrict__ hb) {
    int tid = blockIdx.x * blockDim.x + threadIdx.x;
    if (tid >= N_ * GIN_) return;
    const float v = hb[tid];
    hb[tid] = v > 0.0f ? v : 0.1f * (__expf(v) - 1.0f);
}

// -------------------------------------------------------------- deg -> dis
__global__ void k_dis(const float* __restrict__ deg, float* __restrict__ dis) {
    int i = blockIdx.x * blockDim.x + threadIdx.x;
    if (i >= N_) return;
    const float dg = deg[i];
    dis[i] = dg > 0.0f ? rsqrtf(dg) : 0.0f;
}

// ------------------------------------------ hw = h(N x 256) @ Wg(256 x 32)
// Wg staged in LDS (32 KB) once per block; B operands come from ds_load.
__global__ void k_gemm(const float* __restrict__ hb,
                       const float* __restrict__ Wg,
                       float* __restrict__ hw) {
    __shared__ float sWg[GIN_ * OUT_];         // 8192 floats = 32 KB
    {
        const float4* g4 = (const float4*)Wg;
        float4*       s4 = (float4*)sWg;
        for (int idx = threadIdx.x; idx < (GIN_ * OUT_) / 4; idx += blockDim.x)
            s4[idx] = g4[idx];
    }
    __syncthreads();

    const int wave = threadIdx.x >> 5;
    const int lane = threadIdx.x & 31;
    const int tile = blockIdx.x * (blockDim.x >> 5) + wave;
    if (tile * 16 >= N_) return;              // wave-uniform (after barrier)
    const int base = tile * 16;
    const int half = lane >> 4;
    const int l    = lane & 15;

    v8f acc0 = {}, acc1 = {};
    const float* hrow = hb + (size_t)(base + l) * GIN_;
    for (int k0 = 0; k0 < GIN_; k0 += 4) {
        v2f A;
        A.x = hrow[k0 + 2 * half + 0];
        A.y = hrow[k0 + 2 * half + 1];
        v2f B0, B1;
        B0.x = sWg[(k0 + 2 * half + 0) * OUT_ + l];
        B0.y = sWg[(k0 + 2 * half + 1) * OUT_ + l];
        B1.x = sWg[(k0 + 2 * half + 0) * OUT_ + 16 + l];
        B1.y = sWg[(k0 + 2 * half + 1) * OUT_ + 16 + l];
        acc0 = __builtin_amdgcn_wmma_f32_16x16x4_f32(false, A, false, B0,
                                                     (short)0, acc0, false, false);
        acc1 = __builtin_amdgcn_wmma_f32_16x16x4_f32(false, A, false, B1,
                                                     (short)0, acc1, false, false);
    }
#pragma unroll
    for (int i = 0; i < 8; ++i) {
        const int row = base + i + 8 * half;
        hw[(size_t)row * OUT_ + l]      = acc0[i];
        hw[(size_t)row * OUT_ + 16 + l] = acc1[i];
    }
}

// ------------------------------------- out = bg + self-loop term (norm=2*dis^2)
__global__ void k_outinit(const float* __restrict__ bg,
                          const float* __restrict__ dis,
                          const float* __restrict__ hw,
                          float* __restrict__ out) {
    int tid = blockIdx.x * blockDim.x + threadIdx.x;
    if (tid >= N_ * OUT_) return;
    const int i = tid >> 5;
    const int o = tid & 31;
    const float di = dis[i];
    out[tid] = bg[o] + di * 2.0f * di * hw[tid];
}

// ----------------------------------------- normalized GCN edge scatter
// One 32-lane wave per edge: contiguous hw row read, contiguous atomic row.
__global__ void k_scatter(const int* __restrict__ ei,
                          const float* __restrict__ ea,
                          const float* __restrict__ dis,
                          const float* __restrict__ hw,
                          float* __restrict__ out) {
    int tid = blockIdx.x * blockDim.x + threadIdx.x;
    if (tid >= E_ * OUT_) return;
    const int e = tid >> 5;
    const int o = tid & 31;
    const int s = ei[e];
    const int d = ei[E_ + e];
    const float norm = dis[s] * ea[2 * e + 1] * dis[d];
    atomicAdd(&out[(size_t)d * OUT_ + o], norm * hw[(size_t)s * OUT_ + o]);
}

// =========================================================================
extern "C" void kernel_launch(void* const* d_in, const int* in_sizes, int n_in,
                              void* d_out, int out_size, void* d_ws, size_t ws_size,
                              hipStream_t stream) {
    const float* x   = (const float*)d_in[0];
    const float* ea  = (const float*)d_in[1];
    const float* Wq  = (const float*)d_in[2];
    const float* bq  = (const float*)d_in[3];
    const float* Wk  = (const float*)d_in[4];
    const float* bk  = (const float*)d_in[5];
    const float* Wv  = (const float*)d_in[6];
    const float* bv  = (const float*)d_in[7];
    const float* We  = (const float*)d_in[8];
    const float* Wsk = (const float*)d_in[9];
    const float* bsk = (const float*)d_in[10];
    const float* Wg  = (const float*)d_in[11];
    const float* bg  = (const float*)d_in[12];
    const int*   ei  = (const int*)d_in[13];
    float* out = (float*)d_out;

    // workspace carve-up (256B aligned)
    char* ws = (char*)d_ws;
    size_t off = 0;
    auto carve = [&](size_t bytes) -> float* {
        float* p = (float*)(ws + off);
        off += (bytes + 255) & ~(size_t)255;
        return p;
    };
    float* qb     = carve((size_t)N_ * GIN_ * 4);   // 51.2 MB
    float* kb     = carve((size_t)N_ * GIN_ * 4);   // 51.2 MB
    float* vb     = carve((size_t)N_ * GIN_ * 4);   // 51.2 MB
    float* hb     = carve((size_t)N_ * GIN_ * 4);   // 51.2 MB (skip+agg, then ELU)
    float* logits = carve((size_t)E_ * 16 * 4);     // 102.4 MB (logits, then p)
    float* m      = carve((size_t)N_ * 16 * 4);
    float* z      = carve((size_t)N_ * 16 * 4);
    float* deg    = carve((size_t)N_ * 4);
    float* dis    = carve((size_t)N_ * 4);
    float* hw     = carve((size_t)N_ * OUT_ * 4);

    const int BLK = 256;
    const int tiles = N_ / 16;                       // 3125 (N divisible by 16)
    const int wmmaBlocks = (tiles + 7) / 8;          // 8 waves / block

    k_init<<<(N_ * 16 + BLK - 1) / BLK, BLK, 0, stream>>>(m, z, deg);

    k_proj<<<wmmaBlocks, BLK, 0, stream>>>(x, Wq, bq, Wk, bk, Wv, bv, Wsk, bsk,
                                           qb, kb, vb, hb);

    k_edge_logits<<<(E_ * BANDS_ + BLK - 1) / BLK, BLK, 0, stream>>>(
        ei, ea, We, qb, kb, logits, m, deg);

    k_edge_exp<<<(E_ * 16 + BLK - 1) / BLK, BLK, 0, stream>>>(ei, logits, m, z);

    k_edge_agg<<<(E_ * BANDS_ + BLK - 1) / BLK, BLK, 0, stream>>>(
        ei, ea, We, vb, logits, z, hb);

    k_elu<<<(N_ * GIN_ + BLK - 1) / BLK, BLK, 0, stream>>>(hb);

    k_dis<<<(N_ + BLK - 1) / BLK, BLK, 0, stream>>>(deg, dis);

    k_gemm<<<wmmaBlocks, BLK, 0, stream>>>(hb, Wg, hw);

    k_outinit<<<(N_ * OUT_ + BLK - 1) / BLK, BLK, 0, stream>>>(bg, dis, hw, out);

    k_scatter<<<(E_ * OUT_ + BLK - 1) / BLK, BLK, 0, stream>>>(ei, ea, dis, hw, out);
}